// TA_GNN_60052232732767
// MI455X (gfx1250) — compile-verified
//
#include <hip/hip_runtime.h>
#include <hip/hip_bf16.h>
#include <math.h>

#define H 100
#define H3 300
#define NSESS 1024
#define BATCH 64
#define LSEQ 16
#define NEDGE 2048
#define NNODE 40000
#define KSTEPS 25      // H / 4
#define TPW 10         // node tiles per wave in k_main
#define NTILE 2500     // NNODE / 16

typedef float v2f __attribute__((ext_vector_type(2)));
typedef float v8f __attribute__((ext_vector_type(8)));

__device__ __forceinline__ float sigm(float x) { return 1.0f / (1.0f + __expf(-x)); }

// ---------------- Kernel 1: h0 = embed[x]; m = h0 @ ggc_w; agg = 0 ----------------
__global__ void k_gather(const int* __restrict__ x, const float* __restrict__ embed,
                         const float* __restrict__ ggc, float* __restrict__ h0,
                         float* __restrict__ m, float* __restrict__ agg) {
    int i = blockIdx.x, t = threadIdx.x;
    __shared__ float hrow[H];
    int node = x[i];
    if (t < H) {
        float val = embed[(size_t)node * H + t];
        hrow[t] = val;
        h0[i * H + t] = val;
        agg[i * H + t] = 0.0f;
    }
    __syncthreads();
    if (t < H) {
        float acc = 0.0f;
        for (int k = 0; k < H; ++k) acc = fmaf(hrow[k], ggc[k * H + t], acc);
        m[i * H + t] = acc;
    }
}

// ---------------- Kernel 2: agg[dst] += w_e * m[src] ----------------
__global__ void k_scatter(const int* __restrict__ ei, const float* __restrict__ ew,
                          const float* __restrict__ m, float* __restrict__ agg) {
    int idx = blockIdx.x * blockDim.x + threadIdx.x;
    if (idx >= NEDGE * H) return;
    int e = idx / H, d = idx % H;
    int s = ei[e], dn = ei[NEDGE + e];
    atomicAdd(&agg[dn * H + d], ew[e] * m[s * H + d]);
}

// ---------------- Kernel 3: GRU cell -> v ----------------
__global__ void k_gru(const float* __restrict__ agg, const float* __restrict__ h0,
                      const float* __restrict__ wih, const float* __restrict__ whh,
                      const float* __restrict__ bih, const float* __restrict__ bhh,
                      float* __restrict__ v) {
    int i = blockIdx.x, t = threadIdx.x;
    __shared__ float ar[H], hr[H];
    if (t < H) { ar[t] = agg[i * H + t]; hr[t] = h0[i * H + t]; }
    __syncthreads();
    if (t < H) {
        float ir = bih[t], iz = bih[H + t], in = bih[2 * H + t];
        float hrv = bhh[t], hz = bhh[H + t], hn = bhh[2 * H + t];
        for (int k = 0; k < H; ++k) {
            float a = ar[k], h = hr[k];
            ir  = fmaf(a, wih[t * H + k], ir);
            iz  = fmaf(a, wih[(H + t) * H + k], iz);
            in  = fmaf(a, wih[(2 * H + t) * H + k], in);
            hrv = fmaf(h, whh[t * H + k], hrv);
            hz  = fmaf(h, whh[(H + t) * H + k], hz);
            hn  = fmaf(h, whh[(2 * H + t) * H + k], hn);
        }
        float r = sigm(ir + hrv);
        float z = sigm(iz + hz);
        float n = tanhf(in + r * hn);
        v[i * H + t] = (1.0f - z) * n + z * hr[t];
    }
}

// ---------------- Kernel 4: s_l, t1 = s_l@W1^T + b1 ----------------
__global__ void k_slw1(const float* __restrict__ v, const float* __restrict__ W1w,
                       const float* __restrict__ W1b, float* __restrict__ s_l,
                       float* __restrict__ t1) {
    int b = blockIdx.x, t = threadIdx.x;
    __shared__ float sl[H];
    if (t < H) { sl[t] = v[(b * LSEQ + LSEQ - 1) * H + t]; s_l[b * H + t] = sl[t]; }
    __syncthreads();
    if (t < H) {
        float acc = W1b[t];
        for (int k = 0; k < H; ++k) acc = fmaf(sl[k], W1w[t * H + k], acc);
        t1[b * H + t] = acc;
    }
}

// ---------------- Kernel 5: vt = v@Wt^T+bt ; alpha = sigmoid(t1 + v@W2^T+b2) @ q ----------------
__global__ void k_att(const float* __restrict__ v, const float* __restrict__ t1,
                      const float* __restrict__ W2w, const float* __restrict__ W2b,
                      const float* __restrict__ Wtw, const float* __restrict__ Wtb,
                      const float* __restrict__ qw, const float* __restrict__ qb,
                      float* __restrict__ vt, float* __restrict__ alpha) {
    int n = blockIdx.x, t = threadIdx.x, b = n / LSEQ;
    __shared__ float vrow[H];
    __shared__ float red[128];
    if (t < H) vrow[t] = v[n * H + t];
    __syncthreads();
    float contrib = 0.0f;
    if (t < H) {
        float a2 = W2b[t], avt = Wtb[t];
        for (int k = 0; k < H; ++k) {
            float vk = vrow[k];
            a2  = fmaf(vk, W2w[t * H + k], a2);
            avt = fmaf(vk, Wtw[t * H + k], avt);
        }
        vt[n * H + t] = avt;
        contrib = sigm(t1[b * H + t] + a2) * qw[t];
    }
    red[t] = contrib;
    __syncthreads();
    for (int s = 64; s > 0; s >>= 1) {
        if (t < s) red[t] += red[t + s];
        __syncthreads();
    }
    if (t == 0) alpha[n] = red[0] + qb[0];
}

// ---------------- Kernel 6: s_g = sum alpha*v ; s_h = [s_l,s_g]@W3^T + b3 ----------------
__global__ void k_sgsh(const float* __restrict__ v, const float* __restrict__ alpha,
                       const float* __restrict__ s_l, const float* __restrict__ W3w,
                       const float* __restrict__ W3b, float* __restrict__ s_h) {
    int b = blockIdx.x, t = threadIdx.x;
    __shared__ float sg[H], sl[H];
    if (t < H) {
        float acc = 0.0f;
        for (int l = 0; l < LSEQ; ++l)
            acc = fmaf(alpha[b * LSEQ + l], v[(b * LSEQ + l) * H + t], acc);
        sg[t] = acc;
        sl[t] = s_l[b * H + t];
    }
    __syncthreads();
    if (t < H) {
        float acc = W3b[t];
        for (int k = 0; k < H; ++k) {
            acc = fmaf(sl[k], W3w[t * 2 * H + k], acc);
            acc = fmaf(sg[k], W3w[t * 2 * H + H + k], acc);
        }
        s_h[b * H + t] = acc;
    }
}

// ---------------- Kernel 7 (dominant): fused WMMA logits/C + softmax + s_h.embed ----------------
// One wave per block, one batch b per wave, TPW node tiles per wave.
// B-operand fragments (vt[b], v[b], s_h[b]) are hoisted into registers once and
// reused across all tiles: inner K-step = 1 A-load + 2 WMMA + 2 FMA.
__global__ __launch_bounds__(32) void k_main(const float* __restrict__ embed,
                                             const float* __restrict__ vbuf,
                                             const float* __restrict__ vtbuf,
                                             const float* __restrict__ sh,
                                             float* __restrict__ out) {
    const int lane = threadIdx.x;
    const int b = blockIdx.y;
    const int tile0 = blockIdx.x * TPW;       // 0..2490 step 10
    const int q = lane & 15;
    const int half = lane >> 4;
    const int k0 = half * 2;

    // ---- hoist B fragments (invariant over node tiles) into registers ----
    const float* __restrict__ bprow = vtbuf + (size_t)(b * LSEQ + q) * H + k0;
    const float* __restrict__ bcrow = vbuf  + (size_t)(b * LSEQ + q) * H + k0;
    const float* __restrict__ shrow = sh + b * H + k0;
    v2f bp[KSTEPS], bc[KSTEPS], sv[KSTEPS];
    #pragma unroll
    for (int i = 0; i < KSTEPS; ++i) {
        bp[i] = *(const v2f*)(bprow + 4 * i);
        bc[i] = *(const v2f*)(bcrow + 4 * i);
        sv[i] = *(const v2f*)(shrow + 4 * i);
    }

    __shared__ float lp[256];
    __shared__ float lc[256];
    __shared__ float ls[32];

    for (int t = 0; t < TPW; ++t) {
        const int node0 = (tile0 + t) * 16;
        const float* __restrict__ arow = embed + (size_t)(node0 + q) * H + k0;

        // prefetch next tile's 6.4KB embed slab (uniform guard, EXEC untouched)
        if (tile0 + t + 1 < NTILE) {
            __builtin_prefetch(embed + (size_t)(node0 + 16) * H + lane * 50, 0, 0);
        }

        v8f accp = {};
        v8f accc = {};
        float ssum = 0.0f;
        #pragma unroll
        for (int i = 0; i < KSTEPS; ++i) {
            v2f a = *(const v2f*)(arow + 4 * i);
            accp = __builtin_amdgcn_wmma_f32_16x16x4_f32(false, a, false, bp[i], (short)0, accp, false, false);
            accc = __builtin_amdgcn_wmma_f32_16x16x4_f32(false, a, false, bc[i], (short)0, accc, false, false);
            ssum = fmaf(a.x, sv[i].x, fmaf(a.y, sv[i].y, ssum));
        }

        #pragma unroll
        for (int r = 0; r < 8; ++r) {
            int mrow = r + half * 8;
            lp[mrow * 16 + q] = accp[r];
            lc[mrow * 16 + q] = accc[r];
        }
        ls[lane] = ssum;
        __syncthreads();   // single-wave WG: S_NOP + compiler memory fence

        if (lane < 16) {
            const int mrow = lane;
            float mx = -INFINITY;
            #pragma unroll
            for (int l = 0; l < 16; ++l) mx = fmaxf(mx, lp[mrow * 16 + l]);
            float se = 0.0f, ac = 0.0f;
            #pragma unroll
            for (int l = 0; l < 16; ++l) {
                float e = __expf(lp[mrow * 16 + l] - mx);
                se += e;
                ac = fmaf(e, lc[mrow * 16 + l], ac);
            }
            float shd = ls[mrow] + ls[mrow + 16];
            out[(size_t)b * NNODE + node0 + mrow] = shd + ac / se;
        }
        __syncthreads();   // keep next tile's LDS writes behind this tile's reads
    }
}

extern "C" void kernel_launch(void* const* d_in, const int* in_sizes, int n_in,
                              void* d_out, int out_size, void* d_ws, size_t ws_size,
                              hipStream_t stream) {
    const int*   x     = (const int*)d_in[0];
    const int*   ei    = (const int*)d_in[1];
    const float* ew    = (const float*)d_in[2];
    // d_in[3] = batch (implied by n/16)
    const float* embed = (const float*)d_in[4];
    const float* ggc   = (const float*)d_in[5];
    const float* wih   = (const float*)d_in[6];
    const float* whh   = (const float*)d_in[7];
    const float* bih   = (const float*)d_in[8];
    const float* bhh   = (const float*)d_in[9];
    const float* W1w   = (const float*)d_in[10];
    const float* W1b   = (const float*)d_in[11];
    const float* W2w   = (const float*)d_in[12];
    const float* W2b   = (const float*)d_in[13];
    const float* Wtw   = (const float*)d_in[14];
    const float* Wtb   = (const float*)d_in[15];
    const float* qw    = (const float*)d_in[16];
    const float* qb    = (const float*)d_in[17];
    const float* W3w   = (const float*)d_in[18];
    const float* W3b   = (const float*)d_in[19];
    float* out = (float*)d_out;

    float* ws    = (float*)d_ws;
    float* h0    = ws;                 // 1024*100
    float* m     = ws + 102400;        // 1024*100
    float* agg   = ws + 204800;        // 1024*100
    float* v     = ws + 307200;        // 1024*100
    float* vt    = ws + 409600;        // 1024*100
    float* s_l   = ws + 512000;        // 64*100
    float* t1    = ws + 518400;        // 64*100
    float* s_h   = ws + 524800;        // 64*100
    float* alpha = ws + 531200;        // 1024

    k_gather<<<NSESS, 128, 0, stream>>>(x, embed, ggc, h0, m, agg);
    k_scatter<<<(NEDGE * H + 255) / 256, 256, 0, stream>>>(ei, ew, m, agg);
    k_gru<<<NSESS, 128, 0, stream>>>(agg, h0, wih, whh, bih, bhh, v);
    k_slw1<<<BATCH, 128, 0, stream>>>(v, W1w, W1b, s_l, t1);
    k_att<<<NSESS, 128, 0, stream>>>(v, t1, W2w, W2b, Wtw, Wtb, qw, qb, vt, alpha);
    k_sgsh<<<BATCH, 128, 0, stream>>>(v, alpha, s_l, W3w, W3b, s_h);
    dim3 grid(NTILE / TPW, BATCH);
    k_main<<<grid, 32, 0, stream>>>(embed, v, vt, s_h, out);
}